// MeshDepthFitterEnergy_2551210574744
// MI455X (gfx1250) — compile-verified
//
#include <hip/hip_runtime.h>

// ---------------- problem constants (from reference) ----------------
#define N_VERTS   1000000
#define N_EDGES   3000000
#define IMG_H     1024
#define IMG_W     1024
#define HW        (IMG_H * IMG_W)
#define MAX_DEPTH 10.0f
#define CREGU     2000.0f

// ---------------- workspace layout (units: 4-byte words) ----------------
// [0, HW)                      : depth buffer (uint bits of positive f32)
// [HW, HW+4)                   : mean sums (x,y,z) + pad
// [HW+4, HW+8)                 : energy accumulators [0]=e_data [1]=e_rigid + pad
// [HW+8, HW+8+N_VERTS)         : deg
// [HW+8+N_VERTS, +3*N_VERTS)   : neigh (xyz interleaved)
#define WS_MSUM   (HW)
#define WS_EACC   (HW + 4)
#define WS_DEG    (HW + 8)
#define WS_NEIGH  (HW + 8 + N_VERTS)
#define WS_TOTAL  (HW + 8 + 4 * N_VERTS)   // multiple of 4 words

typedef __attribute__((ext_vector_type(2))) float v2f;
typedef __attribute__((ext_vector_type(8))) float v8f;

// Full-wave (32-lane) sum of two partials per lane, via the CDNA5 matrix unit.
// A = per-lane partials as a 16x4 f32 matrix (2 VGPRs), B = ones (layout-
// independent), D = A x B + 0 -> D[m][n] = rowsum(A,m).  Per the ISA C/D
// layout (VGPR r: lanes 0-15 hold M=r, lanes 16-31 hold M=r+8), summing the
// 8 D registers leaves sum(rows 0..7) in lanes 0-15 and sum(rows 8..15) in
// lanes 16-31; rowsum(m) covers lanes m and m+16, so t + shfl_xor(t,16) is
// the full 64-value wave total in every lane.  EXEC must be all ones here.
__device__ __forceinline__ float wave_sum_wmma(float a0, float a1) {
    v2f A; A[0] = a0;   A[1] = a1;
    v2f B; B[0] = 1.0f; B[1] = 1.0f;
    v8f C = {};
    v8f D = __builtin_amdgcn_wmma_f32_16x16x4_f32(
        /*neg_a=*/false, A, /*neg_b=*/false, B,
        /*c_mod=*/(short)0, C, /*reuse_a=*/false, /*reuse_b=*/false);
    float t = 0.0f;
#pragma unroll
    for (int r = 0; r < 8; ++r) t += D[r];
    t += __shfl_xor(t, 16);
    return t;
}

// Combine per-wave totals across the block in LDS; one atomic per block.
__device__ __forceinline__ void block_accumulate(float wave_total, float* dst) {
    __shared__ float red[32];
    int wave = threadIdx.x >> 5;
    if ((threadIdx.x & 31) == 0) red[wave] = wave_total;
    __syncthreads();
    if (threadIdx.x == 0) {
        float s = 0.0f;
        int nwaves = (int)(blockDim.x >> 5);
        for (int w = 0; w < nwaves; ++w) s += red[w];
        atomicAdd(dst, s);
    }
}

// ---------------- kernel 0: init workspace (b128 stores) ----------------
__global__ void k_init(uint4* __restrict__ wsu4) {
    const uint4 depth_init = make_uint4(0x41200000u, 0x41200000u, 0x41200000u, 0x41200000u);
    const uint4 zero       = make_uint4(0u, 0u, 0u, 0u);
    const int total4 = WS_TOTAL / 4;   // depth region is a whole number of uint4s
    int stride = gridDim.x * blockDim.x;
    for (int i = blockIdx.x * blockDim.x + threadIdx.x; i < total4; i += stride)
        wsu4[i] = (i < HW / 4) ? depth_init : zero;
}

// ---------------- kernel 1: vertex mean (sums) ----------------
__global__ void k_mean(const float* __restrict__ v, float* __restrict__ msum) {
    float sx = 0.f, sy = 0.f, sz = 0.f;
    int stride = gridDim.x * blockDim.x;
    for (int i = blockIdx.x * blockDim.x + threadIdx.x; i < N_VERTS; i += stride) {
        sx += v[3 * i + 0];
        sy += v[3 * i + 1];
        sz += v[3 * i + 2];
    }
#pragma unroll
    for (int o = 16; o > 0; o >>= 1) {
        sx += __shfl_down(sx, o);
        sy += __shfl_down(sy, o);
        sz += __shfl_down(sz, o);
    }
    if ((threadIdx.x & 31) == 0) {
        atomicAdd(&msum[0], sx);
        atomicAdd(&msum[1], sy);
        atomicAdd(&msum[2], sz);
    }
}

// ---------------- kernel 2: transform + project + depth scatter-min ----------------
__global__ void k_project(const float* __restrict__ verts,
                          const float* __restrict__ quat,
                          const float* __restrict__ trans,
                          const float* __restrict__ intr,
                          const float* __restrict__ extr,
                          const float* __restrict__ msum,
                          unsigned* __restrict__ depth) {
    int i = blockIdx.x * blockDim.x + threadIdx.x;
    if (i >= N_VERTS) return;

    float q0 = quat[0], q1 = quat[1], q2 = quat[2], q3 = quat[3];
    float qn = 1.0f / sqrtf(q0 * q0 + q1 * q1 + q2 * q2 + q3 * q3);
    q0 *= qn; q1 *= qn; q2 *= qn; q3 *= qn;

    const float inv_n = 1.0f / (float)N_VERTS;
    float vx = verts[3 * i + 0] - msum[0] * inv_n;
    float vy = verts[3 * i + 1] - msum[1] * inv_n;
    float vz = verts[3 * i + 2] - msum[2] * inv_n;

    // uv = cross(qvec, v); uuv = cross(qvec, uv); v' = v + 2*(w*uv + uuv)
    float ux = q1 * vz - q2 * vy;
    float uy = q2 * vx - q0 * vz;
    float uz = q0 * vy - q1 * vx;
    float wx = q1 * uz - q2 * uy;
    float wy = q2 * ux - q0 * uz;
    float wz = q0 * uy - q1 * ux;
    float tx = vx + 2.0f * (q3 * ux + wx) + trans[0];
    float ty = vy + 2.0f * (q3 * uy + wy) + trans[1];
    float tz = vz + 2.0f * (q3 * uz + wz) + trans[2];

    // p_cam = v_t @ E[:, :3]^T + E[:, 3]   (E row-major 3x4)
    float px = extr[0] * tx + extr[1] * ty + extr[2]  * tz + extr[3];
    float py = extr[4] * tx + extr[5] * ty + extr[6]  * tz + extr[7];
    float pz = extr[8] * tx + extr[9] * ty + extr[10] * tz + extr[11];

    // proj = p_cam @ K^T   (K row-major 3x3)
    float pr0 = intr[0] * px + intr[1] * py + intr[2] * pz;
    float pr1 = intr[3] * px + intr[4] * py + intr[5] * pz;
    float pr2 = intr[6] * px + intr[7] * py + intr[8] * pz;

    float u = pr0 / pr2;
    float vv = pr1 / pr2;
    int xi = (int)fminf(fmaxf(rintf(u), 0.0f), (float)(IMG_W - 1));
    int yi = (int)fminf(fmaxf(rintf(vv), 0.0f), (float)(IMG_H - 1));

    float z = pz; // DEPTH_SCALE == 1
    float zvalid = (z > 0.0f) ? z : MAX_DEPTH;
    // all candidates are positive floats -> u32 bit pattern preserves order
    atomicMin(&depth[yi * IMG_W + xi], __float_as_uint(zvalid));
}

// ---------------- kernel 3: e_data = sum((clip(depth) - hand)^2) ----------------
// Launched with exactly HW/8 threads: each thread consumes two float4s
// (one into each WMMA accumulator slot). b128 loads, no loop, no tail.
__device__ __forceinline__ float sq4(uint4 d, float4 h) {
    float acc = 0.0f;
    float e0 = fminf(fmaxf(__uint_as_float(d.x), 0.0f), MAX_DEPTH) - h.x;
    float e1 = fminf(fmaxf(__uint_as_float(d.y), 0.0f), MAX_DEPTH) - h.y;
    float e2 = fminf(fmaxf(__uint_as_float(d.z), 0.0f), MAX_DEPTH) - h.z;
    float e3 = fminf(fmaxf(__uint_as_float(d.w), 0.0f), MAX_DEPTH) - h.w;
    acc = e0 * e0 + e1 * e1 + e2 * e2 + e3 * e3;
    return acc;
}

__global__ void k_edata(const uint4* __restrict__ depth4,
                        const float4* __restrict__ hand4,
                        float* __restrict__ eacc) {
    int tid = blockIdx.x * blockDim.x + threadIdx.x;     // 0 .. HW/8-1
    int stride = gridDim.x * blockDim.x;                 // == HW/8
    float a0 = sq4(depth4[tid],          hand4[tid]);
    float a1 = sq4(depth4[tid + stride], hand4[tid + stride]);
    float t = wave_sum_wmma(a0, a1);   // EXEC all ones (uniform path)
    block_accumulate(t, &eacc[0]);
}

// ---------------- kernel 4: edge scatter (deg, neigh) ----------------
__global__ void k_edge(const float* __restrict__ verts,
                       const float* __restrict__ vref,
                       const int* __restrict__ edges,
                       float* __restrict__ deg,
                       float* __restrict__ neigh) {
    int e = blockIdx.x * blockDim.x + threadIdx.x;
    if (e >= N_EDGES) return;
    int s = edges[2 * e + 0];
    int d = edges[2 * e + 1];
    float dx = verts[3 * d + 0] - vref[3 * d + 0];
    float dy = verts[3 * d + 1] - vref[3 * d + 1];
    float dz = verts[3 * d + 2] - vref[3 * d + 2];
    atomicAdd(&deg[s], 1.0f);
    atomicAdd(&neigh[3 * s + 0], dx);
    atomicAdd(&neigh[3 * s + 1], dy);
    atomicAdd(&neigh[3 * s + 2], dz);
}

// ---------------- kernel 5: e_rigid = sum((deg*dv - neigh)^2) ----------------
__device__ __forceinline__ float rigid_contrib(int i,
                                               const float* __restrict__ verts,
                                               const float* __restrict__ vref,
                                               const float* __restrict__ deg,
                                               const float* __restrict__ neigh) {
    float dg = deg[i];
    float lx = dg * (verts[3 * i + 0] - vref[3 * i + 0]) - neigh[3 * i + 0];
    float ly = dg * (verts[3 * i + 1] - vref[3 * i + 1]) - neigh[3 * i + 1];
    float lz = dg * (verts[3 * i + 2] - vref[3 * i + 2]) - neigh[3 * i + 2];
    return lx * lx + ly * ly + lz * lz;
}

__global__ void k_erigid(const float* __restrict__ verts,
                         const float* __restrict__ vref,
                         const float* __restrict__ deg,
                         const float* __restrict__ neigh,
                         float* __restrict__ eacc) {
    float a0 = 0.f, a1 = 0.f;
    int tid = blockIdx.x * blockDim.x + threadIdx.x;
    int stride = gridDim.x * blockDim.x;
    for (int i = tid; i < N_VERTS; i += 2 * stride) {
        a0 += rigid_contrib(i, verts, vref, deg, neigh);
        int j = i + stride;
        if (j < N_VERTS) a1 += rigid_contrib(j, verts, vref, deg, neigh);
    }
    float t = wave_sum_wmma(a0, a1);   // EXEC all ones (post-loop, uniform)
    block_accumulate(t, &eacc[1]);
}

// ---------------- kernel 6: finalize ----------------
__global__ void k_final(const float* __restrict__ eacc, float* __restrict__ out) {
    if (blockIdx.x == 0 && threadIdx.x == 0)
        out[0] = eacc[0] + CREGU * eacc[1];
}

// ---------------- host launcher ----------------
extern "C" void kernel_launch(void* const* d_in, const int* in_sizes, int n_in,
                              void* d_out, int out_size, void* d_ws, size_t ws_size,
                              hipStream_t stream) {
    const float* verts = (const float*)d_in[0];  // (N_VERTS,3)
    const float* vref  = (const float*)d_in[1];  // (N_VERTS,3)
    const float* quat  = (const float*)d_in[2];  // (4,)
    const float* trans = (const float*)d_in[3];  // (3,)
    const float* hand  = (const float*)d_in[4];  // (H,W)
    const float* intr  = (const float*)d_in[5];  // (3,3)
    const float* extr  = (const float*)d_in[6];  // (3,4)
    const int*   edges = (const int*)d_in[7];    // (N_EDGES,2) int32
    float* out = (float*)d_out;

    float*    wsf = (float*)d_ws;
    unsigned* depth = (unsigned*)d_ws;
    float* msum  = wsf + WS_MSUM;
    float* eacc  = wsf + WS_EACC;
    float* deg   = wsf + WS_DEG;
    float* neigh = wsf + WS_NEIGH;

    const int B = 256;

    k_init<<<2048, B, 0, stream>>>((uint4*)d_ws);
    k_mean<<<1024, B, 0, stream>>>(verts, msum);
    k_project<<<(N_VERTS + B - 1) / B, B, 0, stream>>>(verts, quat, trans, intr,
                                                       extr, msum, depth);
    // HW/8 threads total: each handles exactly two float4s
    k_edata<<<(HW / 8) / B, B, 0, stream>>>((const uint4*)depth,
                                            (const float4*)hand, eacc);
    k_edge<<<(N_EDGES + B - 1) / B, B, 0, stream>>>(verts, vref, edges, deg, neigh);
    k_erigid<<<2048, B, 0, stream>>>(verts, vref, deg, neigh, eacc);
    k_final<<<1, 32, 0, stream>>>(eacc, out);
}